// EdgeSAGEConv_26053271617546
// MI455X (gfx1250) — compile-verified
//
#include <hip/hip_runtime.h>
#include <hip/hip_bf16.h>

// ---------------------------------------------------------------------------
// EdgeSAGEConv for MI455X (gfx1250, wave32)
//   CSR-by-dst build (int atomics only) -> register-resident gather-mean
//   -> 64-row-panel f32 WMMA GEMM (V_WMMA_F32_16X16X4_F32, 4x B reuse)
// ---------------------------------------------------------------------------

#define IN_FEATS   128
#define EDGE_FEATS 32
#define OUT_FEATS  128
#define K_TOTAL    (IN_FEATS + EDGE_FEATS)   // 160 (h_neigh width)
#define LDA        292                       // 288 + 4 pad: conflict-free LDS
#define MROWS      64                        // M panel rows (4 x 16 tiles)

typedef float v2f __attribute__((ext_vector_type(2)));
typedef float v8f __attribute__((ext_vector_type(8)));

// ---------------------------------------------------------------------------
__global__ void zero_i32_kernel(int* __restrict__ p, int n) {
    int i = blockIdx.x * blockDim.x + threadIdx.x;
    int stride = gridDim.x * blockDim.x;
    for (; i < n; i += stride) p[i] = 0;
}

// degree histogram (int atomics: 1.6M total)
__global__ void hist_kernel(const int* __restrict__ dst, int* __restrict__ degcnt,
                            int n_edges) {
    int e = blockIdx.x * blockDim.x + threadIdx.x;
    int stride = gridDim.x * blockDim.x;
    for (; e < n_edges; e += stride) atomicAdd(&degcnt[dst[e]], 1);
}

// per-block exclusive scan (Hillis-Steele in LDS), block totals out
__global__ __launch_bounds__(256)
void scan_block_kernel(const int* __restrict__ degcnt, int* __restrict__ rowstart,
                       int* __restrict__ blocksums, int n) {
    __shared__ int s[256];
    const int tid = threadIdx.x;
    const int i = blockIdx.x * 256 + tid;
    const int v = (i < n) ? degcnt[i] : 0;
    s[tid] = v;
    __syncthreads();
    for (int off = 1; off < 256; off <<= 1) {
        const int t = (tid >= off) ? s[tid - off] : 0;
        __syncthreads();
        s[tid] += t;
        __syncthreads();
    }
    if (i < n) rowstart[i] = s[tid] - v;     // block-local exclusive
    if (tid == 255) blocksums[blockIdx.x] = s[255];
}

// serial exclusive scan of the (<=1024) block totals
__global__ void scan_sums_kernel(int* __restrict__ blocksums, int nblocks) {
    if (blockIdx.x == 0 && threadIdx.x == 0) {
        int run = 0;
        for (int b = 0; b < nblocks; ++b) {
            const int t = blocksums[b];
            blocksums[b] = run;
            run += t;
        }
    }
}

// add block offsets; initialize bin cursors; cap rowstart[n]
__global__ __launch_bounds__(256)
void scan_add_kernel(int* __restrict__ rowstart, int* __restrict__ cursor,
                     const int* __restrict__ blocksums, int n, int n_edges) {
    const int i = blockIdx.x * 256 + threadIdx.x;
    if (i < n) {
        const int r = rowstart[i] + blocksums[blockIdx.x];
        rowstart[i] = r;
        cursor[i] = r;
    }
    if (i == 0) rowstart[n] = n_edges;
}

// bucket edge ids by dst (1.6M int atomics with return)
__global__ void bin_edges_kernel(const int* __restrict__ dst, int* __restrict__ cursor,
                                 int* __restrict__ edge_ids, int n_edges) {
    int e = blockIdx.x * blockDim.x + threadIdx.x;
    int stride = gridDim.x * blockDim.x;
    for (; e < n_edges; e += stride) {
        const int pos = atomicAdd(&cursor[dst[e]], 1);
        edge_ids[pos] = e;
    }
}

// ---------------------------------------------------------------------------
// One wave per node: register-accumulated neighborhood mean, no float atomics.
// Lane owns feat cols [4*lane, 4*lane+4) (float4) and edge col `lane`.
// hmean row layout: [128 mean feat | 32 mean edge], pre-scaled by 1/max(deg,1).
__global__ __launch_bounds__(256)
void gather_mean_kernel(const float* __restrict__ feat,
                        const float* __restrict__ edge_feats,
                        const int* __restrict__ src,
                        const int* __restrict__ rowstart,
                        const int* __restrict__ edge_ids,
                        float* __restrict__ hmean,
                        int n_nodes) {
    const int lane   = threadIdx.x & 31;
    const int wave   = blockIdx.x * (blockDim.x >> 5) + (threadIdx.x >> 5);
    const int nwaves = gridDim.x * (blockDim.x >> 5);

    for (int v = wave; v < n_nodes; v += nwaves) {
        const int beg = rowstart[v];
        const int end = rowstart[v + 1];
        float ax = 0.f, ay = 0.f, az = 0.f, aw = 0.f, ae = 0.f;
        for (int idx = beg; idx < end; ++idx) {
            const int e = edge_ids[idx];
            const int s = src[e];
            const float4 f = ((const float4*)(feat + (size_t)s * IN_FEATS))[lane];
            ax += f.x; ay += f.y; az += f.z; aw += f.w;
            ae += edge_feats[(size_t)e * EDGE_FEATS + lane];
        }
        const float inv = (end > beg) ? 1.0f / (float)(end - beg) : 0.0f;
        float* __restrict__ hv = hmean + (size_t)v * K_TOTAL;
        float4 o; o.x = ax * inv; o.y = ay * inv; o.z = az * inv; o.w = aw * inv;
        ((float4*)hv)[lane] = o;
        hv[IN_FEATS + lane] = ae * inv;
    }
}

// ---------------------------------------------------------------------------
// 64-node M panel in LDS (64 x 292 f32 = 73 KB). 8 waves, each owns a 16-wide
// N slice and 4 M tiles: every B fragment (2 global loads) feeds 4 WMMAs.
__global__ __launch_bounds__(256)
void gemm_wmma_kernel(const float* __restrict__ feat,
                      const float* __restrict__ hmean,
                      const float* __restrict__ Wself,    // [128][128] (k, n)
                      const float* __restrict__ Wneigh,   // [160][128]
                      const float* __restrict__ bias,     // [128]
                      float* __restrict__ out,            // [N][128]
                      int n_nodes) {
    __shared__ float As[MROWS * LDA];

    const int tid   = threadIdx.x;
    const int node0 = blockIdx.x * MROWS;

    // stage feat part: columns [0, 128)
    for (int idx = tid; idx < MROWS * IN_FEATS; idx += 256) {
        const int m = idx >> 7;
        const int c = idx & (IN_FEATS - 1);
        const int node = node0 + m;
        As[m * LDA + c] = (node < n_nodes) ? feat[(size_t)node * IN_FEATS + c] : 0.0f;
    }
    // stage pre-scaled mean part: columns [128, 288)
    for (int idx = tid; idx < MROWS * K_TOTAL; idx += 256) {
        const int m = idx / K_TOTAL;
        const int c = idx % K_TOTAL;
        const int node = node0 + m;
        As[m * LDA + IN_FEATS + c] =
            (node < n_nodes) ? hmean[(size_t)node * K_TOTAL + c] : 0.0f;
    }
    __syncthreads();

    const int lane  = tid & 31;
    const int wave  = tid >> 5;          // 0..7 -> N slice [wave*16, wave*16+16)
    const int mrow  = lane & 15;         // A frag: M = lane % 16
    const int khalf = (lane >> 4) << 1;  // A/B frag: K pair 0/1 or 2/3
    const int ncol  = wave * 16 + (lane & 15);

    v8f acc[4] = {v8f{}, v8f{}, v8f{}, v8f{}};

    // K range 1: feat @ W_self (A columns 0..127)
    for (int k = 0; k < IN_FEATS; k += 4) {
        const float* __restrict__ wb = Wself + (size_t)(k + khalf) * OUT_FEATS + ncol;
        v2f b; b.x = wb[0]; b.y = wb[OUT_FEATS];
#pragma unroll
        for (int t = 0; t < 4; ++t) {
            const float2 av = *(const float2*)&As[(t * 16 + mrow) * LDA + k + khalf];
            v2f a; a.x = av.x; a.y = av.y;
            acc[t] = __builtin_amdgcn_wmma_f32_16x16x4_f32(false, a, false, b,
                                                           (short)0, acc[t],
                                                           false, false);
        }
    }
    // K range 2: h_neigh @ W_neigh (A columns 128..287)
    for (int k = 0; k < K_TOTAL; k += 4) {
        const float* __restrict__ wb = Wneigh + (size_t)(k + khalf) * OUT_FEATS + ncol;
        v2f b; b.x = wb[0]; b.y = wb[OUT_FEATS];
#pragma unroll
        for (int t = 0; t < 4; ++t) {
            const float2 av =
                *(const float2*)&As[(t * 16 + mrow) * LDA + IN_FEATS + k + khalf];
            v2f a; a.x = av.x; a.y = av.y;
            acc[t] = __builtin_amdgcn_wmma_f32_16x16x4_f32(false, a, false, b,
                                                           (short)0, acc[t],
                                                           false, false);
        }
    }

    // D layout (16x16 f32): VGPR r -> M = r + 8*(lane>=16), N = lane%16
    const int mbase = (lane >> 4) * 8;
    const float bv = bias[ncol];
#pragma unroll
    for (int t = 0; t < 4; ++t) {
#pragma unroll
        for (int r = 0; r < 8; ++r) {
            const int node = node0 + t * 16 + mbase + r;
            if (node < n_nodes)
                out[(size_t)node * OUT_FEATS + ncol] = acc[t][r] + bv;
        }
    }
}

// ---------------------------------------------------------------------------
extern "C" void kernel_launch(void* const* d_in, const int* in_sizes, int n_in,
                              void* d_out, int out_size, void* d_ws, size_t ws_size,
                              hipStream_t stream) {
    const float* feat       = (const float*)d_in[0];
    const float* edge_feats = (const float*)d_in[1];
    const int*   src        = (const int*)d_in[2];
    const int*   dst        = (const int*)d_in[3];
    const float* Wself      = (const float*)d_in[4];
    const float* Wneigh     = (const float*)d_in[5];
    const float* bias       = (const float*)d_in[6];
    float*       out        = (float*)d_out;

    const int n_nodes = in_sizes[0] / IN_FEATS;
    const int n_edges = in_sizes[2];

    // workspace layout
    float* hmean    = (float*)d_ws;                    // N*160 f32
    int*   degcnt   = (int*)(hmean + (size_t)n_nodes * K_TOTAL);  // N
    int*   rowstart = degcnt + n_nodes;                // N+1
    int*   cursor   = rowstart + (n_nodes + 1);        // N
    int*   blocksums= cursor + n_nodes;                // <=1024
    int*   edge_ids = blocksums + 1024;                // E

    const int nscan = (n_nodes + 255) / 256;

    zero_i32_kernel<<<512, 256, 0, stream>>>(degcnt, n_nodes);
    hist_kernel<<<2048, 256, 0, stream>>>(dst, degcnt, n_edges);
    scan_block_kernel<<<nscan, 256, 0, stream>>>(degcnt, rowstart, blocksums, n_nodes);
    scan_sums_kernel<<<1, 32, 0, stream>>>(blocksums, nscan);
    scan_add_kernel<<<nscan, 256, 0, stream>>>(rowstart, cursor, blocksums,
                                               n_nodes, n_edges);
    bin_edges_kernel<<<2048, 256, 0, stream>>>(dst, cursor, edge_ids, n_edges);

    const int gather_blocks = (n_nodes + 7) / 8;       // 8 waves per block
    gather_mean_kernel<<<gather_blocks, 256, 0, stream>>>(
        feat, edge_feats, src, rowstart, edge_ids, hmean, n_nodes);

    const int mtiles = (n_nodes + MROWS - 1) / MROWS;
    gemm_wmma_kernel<<<mtiles, 256, 0, stream>>>(
        feat, hmean, Wself, Wneigh, bias, out, n_nodes);
}